// BAQuantizer_81535659147902
// MI455X (gfx1250) — compile-verified
//
#include <hip/hip_runtime.h>
#include <hip/hip_bf16.h>

// Problem constants
#define NVEC   32768      // B*H*W = 32*32*32
#define KCB    1024       // codebook entries
#define DIM    64         // embedding dim
#define EPSQ   1e-10f
#define ZQ_ELEMS 2097152  // 32*64*32*32

typedef float v2f __attribute__((ext_vector_type(2)));
typedef float v8f __attribute__((ext_vector_type(8)));

// ---------------------------------------------------------------------------
// znorm[n] = sum_d z_flat[n,d]^2 ; z is [B=32, D=64, H=32, W=32]
// z_flat[n,d] = z[b*65536 + d*1024 + hw],  n = b*1024 + hw
// ---------------------------------------------------------------------------
__global__ __launch_bounds__(256) void znorm_kernel(const float* __restrict__ z,
                                                    float* __restrict__ znorm) {
    int n = blockIdx.x * 256 + threadIdx.x;          // 32768 threads
    int b = n >> 10, hw = n & 1023;
    const float* zb = z + b * 65536 + hw;
    float s = 0.f;
#pragma unroll
    for (int d = 0; d < DIM; ++d) { float v = zb[d * 1024]; s += v * v; }
    znorm[n] = s;
}

__global__ __launch_bounds__(256) void enorm_kernel(const float* __restrict__ emb,
                                                    float* __restrict__ enorm) {
    int k = blockIdx.x * 256 + threadIdx.x;          // 1024 threads
    const float4* e = (const float4*)(emb + k * DIM);
    float s = 0.f;
#pragma unroll
    for (int j = 0; j < DIM / 4; ++j) {
        float4 v = e[j];
        s += v.x * v.x + v.y * v.y + v.z * v.z + v.w * v.w;
    }
    enorm[k] = s;
}

// logQ = log(max(pi, eps)); zero Qaccum and loss accumulators
__global__ __launch_bounds__(1024) void init_kernel(const float* __restrict__ pi,
                                                    float* __restrict__ logQ,
                                                    float* __restrict__ Qacc,
                                                    float* __restrict__ accum) {
    int k = threadIdx.x;
    logQ[k] = logf(fmaxf(pi[k], EPSQ));
    Qacc[k] = 0.f;
    if (k < 8) accum[k] = 0.f;
}

// ---------------------------------------------------------------------------
// distances[n,k] = znorm[n] + enorm[k] - 2 * (z_flat @ emb^T)[n,k]
// WMMA f32 16x16x4.  One wave -> 16 rows x 64 cols strip (4 accumulators).
// A (16x4, M=n,K=d): lane<16: M=lane, {K0,K1}; lane>=16: M=lane-16, {K2,K3}
// B (4x16, K=d,N=k): vgpr0 holds rows K0(lo)/K2(hi), vgpr1 rows K1(lo)/K3(hi)
// C/D: vgpr r -> rows r (lanes 0-15) and r+8 (lanes 16-31), N = lane&15
// ---------------------------------------------------------------------------
__global__ __launch_bounds__(256) void dist_kernel(const float* __restrict__ z,
                                                   const float* __restrict__ emb,
                                                   const float* __restrict__ znorm,
                                                   const float* __restrict__ enorm,
                                                   float* __restrict__ dist) {
    int wave = blockIdx.x * 8 + (threadIdx.x >> 5);  // 32768 waves
    int lane = threadIdx.x & 31;
    int mtile = wave >> 4;                 // 0..2047 -> n0 = mtile*16
    int kgrp  = wave & 15;                 // 0..15   -> k0 = kgrp*64
    int n0 = mtile << 4, k0 = kgrp << 6;
    int b = n0 >> 10, hw0 = n0 & 1023;     // all 16 rows share b (16 | 1024)
    const float* zb = z + b * 65536 + hw0; // z_flat[n0+m, d] = zb[d*1024 + m]
    bool hi = lane >= 16;
    int l15 = lane & 15;

    v8f c0 = {}, c1 = {}, c2 = {}, c3 = {};
#pragma unroll
    for (int d0 = 0; d0 < DIM; d0 += 4) {
        int ad = d0 + (hi ? 2 : 0);
        v2f a;
        a.x = zb[(ad + 0) * 1024 + l15];
        a.y = zb[(ad + 1) * 1024 + l15];
        const float* e0 = emb + (k0 +  0 + l15) * DIM + ad;
        const float* e1 = emb + (k0 + 16 + l15) * DIM + ad;
        const float* e2 = emb + (k0 + 32 + l15) * DIM + ad;
        const float* e3 = emb + (k0 + 48 + l15) * DIM + ad;
        v2f b0, b1, b2, b3;
        b0.x = e0[0]; b0.y = e0[1];
        b1.x = e1[0]; b1.y = e1[1];
        b2.x = e2[0]; b2.y = e2[1];
        b3.x = e3[0]; b3.y = e3[1];
        c0 = __builtin_amdgcn_wmma_f32_16x16x4_f32(false, a, false, b0, (short)0, c0, false, false);
        c1 = __builtin_amdgcn_wmma_f32_16x16x4_f32(false, a, false, b1, (short)0, c1, false, false);
        c2 = __builtin_amdgcn_wmma_f32_16x16x4_f32(false, a, false, b2, (short)0, c2, false, false);
        c3 = __builtin_amdgcn_wmma_f32_16x16x4_f32(false, a, false, b3, (short)0, c3, false, false);
    }
    float en0 = enorm[k0 +  0 + l15];
    float en1 = enorm[k0 + 16 + l15];
    float en2 = enorm[k0 + 32 + l15];
    float en3 = enorm[k0 + 48 + l15];
#pragma unroll
    for (int r = 0; r < 8; ++r) {
        int m = r + (hi ? 8 : 0);
        float zn = znorm[n0 + m];
        float* row = dist + (size_t)(n0 + m) * KCB + k0;
        row[ 0 + l15] = zn + en0 - 2.f * c0[r];
        row[16 + l15] = zn + en1 - 2.f * c1[r];
        row[32 + l15] = zn + en2 - 2.f * c2[r];
        row[48 + l15] = zn + en3 - 2.f * c3[r];
    }
}

// ---------------------------------------------------------------------------
// One BA iteration: P = softmax(logQ - beta*dist, axis=K); Qacc += sum_n P.
// Wave handles one row at a time: lane owns k = lane + 32j (conflict-free LDS).
// On the last iteration also write P in-place over dist and argmax indices.
// ---------------------------------------------------------------------------
__global__ __launch_bounds__(256) void ba_kernel(float* __restrict__ distP,
                                                 const float* __restrict__ logQ,
                                                 float* __restrict__ Qacc,
                                                 int* __restrict__ idx_out,
                                                 const float* __restrict__ beta_p,
                                                 int writeP) {
    __shared__ float lq[KCB];
    __shared__ float qp[8][KCB];
    int t = threadIdx.x, w = t >> 5, lane = t & 31;
    for (int i = t; i < KCB; i += 256) lq[i] = logQ[i];
    for (int i = t; i < 8 * KCB; i += 256) ((float*)qp)[i] = 0.f;
    __syncthreads();

    float beta = beta_p[0];
    float* qpw = qp[w];
#pragma unroll 1
    for (int rr = 0; rr < 16; ++rr) {               // 128 rows per block
        int n = blockIdx.x * 128 + w * 16 + rr;
        float* row = distP + (size_t)n * KCB;
        float v[32];
        float lm = -3.4e38f, bv = -3.4e38f;
        int bi = 0;
#pragma unroll
        for (int j = 0; j < 32; ++j) {
            int k = lane + (j << 5);
            float lg = lq[k] - beta * row[k];
            v[j] = lg;
            if (lg > bv) { bv = lg; bi = k; }
            lm = fmaxf(lm, lg);
        }
#pragma unroll
        for (int off = 16; off; off >>= 1) lm = fmaxf(lm, __shfl_xor(lm, off, 32));
        float s = 0.f;
#pragma unroll
        for (int j = 0; j < 32; ++j) { float e = __expf(v[j] - lm); v[j] = e; s += e; }
#pragma unroll
        for (int off = 16; off; off >>= 1) s += __shfl_xor(s, off, 32);
        float inv = 1.f / s;
#pragma unroll
        for (int j = 0; j < 32; ++j) {
            int k = lane + (j << 5);
            float p = v[j] * inv;
            qpw[k] += p;
            if (writeP) row[k] = p;                 // safe: row fully read above
        }
        if (writeP) {
#pragma unroll
            for (int off = 16; off; off >>= 1) {
                float ov = __shfl_xor(bv, off, 32);
                int   oi = __shfl_xor(bi, off, 32);
                if (ov > bv || (ov == bv && oi < bi)) { bv = ov; bi = oi; }
            }
            if (lane == 0) idx_out[n] = bi;
        }
    }
    __syncthreads();
    for (int k = t; k < KCB; k += 256) {
        float s2 = 0.f;
#pragma unroll
        for (int w2 = 0; w2 < 8; ++w2) s2 += qp[w2][k];
        atomicAdd(&Qacc[k], s2);
    }
}

// Q = clip(mean(P), eps); logQ = log(Q); reset Qacc; final: accum[1]+=Q*logQ
__global__ __launch_bounds__(1024) void qnorm_kernel(float* __restrict__ Qacc,
                                                     float* __restrict__ logQ,
                                                     float* __restrict__ accum,
                                                     int final_it) {
    int k = threadIdx.x;
    float q = fmaxf(Qacc[k] * (1.0f / (float)NVEC), EPSQ);
    float lg = logf(q);
    logQ[k] = lg;
    Qacc[k] = 0.f;
    if (final_it) atomicAdd(&accum[1], q * lg);
}

// ---------------------------------------------------------------------------
// z_q_soft = P @ emb  (only needed for MSE losses).  WMMA f32 16x16x4.
// One wave -> 16 rows x all 64 dims, K-loop over 1024 codes. Fused MSE reduce.
// ---------------------------------------------------------------------------
__global__ __launch_bounds__(256) void zq_kernel(const float* __restrict__ P,
                                                 const float* __restrict__ emb,
                                                 const float* __restrict__ z,
                                                 float* __restrict__ accum) {
    int wave = blockIdx.x * 8 + (threadIdx.x >> 5); // 2048 waves
    int lane = threadIdx.x & 31;
    int n0 = wave << 4;
    bool hi = lane >= 16;
    int l15 = lane & 15;
    int b = n0 >> 10, hw0 = n0 & 1023;
    const float* zb = z + b * 65536 + hw0;

    v8f c0 = {}, c1 = {}, c2 = {}, c3 = {};
#pragma unroll 4
    for (int ks = 0; ks < KCB; ks += 4) {
        int ak = ks + (hi ? 2 : 0);
        v2f a;
        const float* pr = P + (size_t)(n0 + l15) * KCB + ak;
        a.x = pr[0]; a.y = pr[1];
        const float* er = emb + ak * DIM + l15;
        v2f b0, b1, b2, b3;
        b0.x = er[ 0]; b0.y = er[ 0 + DIM];
        b1.x = er[16]; b1.y = er[16 + DIM];
        b2.x = er[32]; b2.y = er[32 + DIM];
        b3.x = er[48]; b3.y = er[48 + DIM];
        c0 = __builtin_amdgcn_wmma_f32_16x16x4_f32(false, a, false, b0, (short)0, c0, false, false);
        c1 = __builtin_amdgcn_wmma_f32_16x16x4_f32(false, a, false, b1, (short)0, c1, false, false);
        c2 = __builtin_amdgcn_wmma_f32_16x16x4_f32(false, a, false, b2, (short)0, c2, false, false);
        c3 = __builtin_amdgcn_wmma_f32_16x16x4_f32(false, a, false, b3, (short)0, c3, false, false);
    }
    // fused MSE: sum (z_flat - z_q_soft)^2
    float mse = 0.f;
#pragma unroll
    for (int r = 0; r < 8; ++r) {
        int m = r + (hi ? 8 : 0);
        float d0 = zb[( 0 + l15) * 1024 + m] - c0[r];
        float d1 = zb[(16 + l15) * 1024 + m] - c1[r];
        float d2 = zb[(32 + l15) * 1024 + m] - c2[r];
        float d3 = zb[(48 + l15) * 1024 + m] - c3[r];
        mse += d0 * d0 + d1 * d1 + d2 * d2 + d3 * d3;
    }
#pragma unroll
    for (int off = 16; off; off >>= 1) mse += __shfl_xor(mse, off, 32);
    if (lane == 0) atomicAdd(&accum[0], mse);
}

// z_q forward value == emb[argmax] gathered, laid out as [B, D, H, W]
__global__ __launch_bounds__(256) void gather_kernel(const int* __restrict__ idx,
                                                     const float* __restrict__ emb,
                                                     float* __restrict__ out) {
    int e = blockIdx.x * 256 + threadIdx.x;          // 2097152 threads
    int b = e >> 16, rem = e & 65535;
    int d = rem >> 10, hw = rem & 1023;
    int n = (b << 10) + hw;
    out[e] = emb[idx[n] * DIM + d];
}

__global__ __launch_bounds__(256) void idxout_kernel(const int* __restrict__ idx,
                                                     float* __restrict__ out) {
    int n = blockIdx.x * 256 + threadIdx.x;          // 32768 threads
    out[ZQ_ELEMS + 3 + n] = (float)idx[n];
}

__global__ void scalars_kernel(const float* __restrict__ accum,
                               float* __restrict__ out) {
    float mse = accum[0] * (1.0f / (float)(NVEC * DIM));
    out[ZQ_ELEMS + 0] = 0.25f * mse;                 // commitment
    out[ZQ_ELEMS + 1] = mse;                         // codebook
    out[ZQ_ELEMS + 2] = 0.01f * accum[1];            // -0.01 * (-sum Q logQ)
}

extern "C" void kernel_launch(void* const* d_in, const int* in_sizes, int n_in,
                              void* d_out, int out_size, void* d_ws, size_t ws_size,
                              hipStream_t stream) {
    const float* z      = (const float*)d_in[0];     // [32,64,32,32]
    const float* emb    = (const float*)d_in[1];     // [1024,64]
    const float* pi     = (const float*)d_in[2];     // [1024]
    const float* beta_p = (const float*)d_in[3];     // scalar
    float* out = (float*)d_out;

    // workspace layout (floats): P/dist 128MB fits L2 alongside side arrays
    float* P     = (float*)d_ws;                     // 33554432
    float* znorm = P + (size_t)NVEC * KCB;           // 32768
    float* enorm = znorm + NVEC;                     // 1024
    float* logQ  = enorm + KCB;                      // 1024
    float* Qacc  = logQ + KCB;                       // 1024
    float* accum = Qacc + KCB;                       // 8
    int*   idx   = (int*)(accum + 8);                // 32768

    znorm_kernel<<<NVEC / 256, 256, 0, stream>>>(z, znorm);
    enorm_kernel<<<KCB / 256, 256, 0, stream>>>(emb, enorm);
    init_kernel<<<1, 1024, 0, stream>>>(pi, logQ, Qacc, accum);

    dist_kernel<<<(NVEC / 16) * (KCB / 64) / 8, 256, 0, stream>>>(z, emb, znorm, enorm, P);

    for (int it = 0; it < 5; ++it) {
        int last = (it == 4);
        ba_kernel<<<NVEC / 128, 256, 0, stream>>>(P, logQ, Qacc, idx, beta_p, last);
        qnorm_kernel<<<1, 1024, 0, stream>>>(Qacc, logQ, accum, last);
    }

    zq_kernel<<<(NVEC / 16) / 8, 256, 0, stream>>>(P, emb, z, accum);
    gather_kernel<<<ZQ_ELEMS / 256, 256, 0, stream>>>(idx, emb, out);
    idxout_kernel<<<NVEC / 256, 256, 0, stream>>>(idx, out);
    scalars_kernel<<<1, 1, 0, stream>>>(accum, out);
}